// GCNLayerWithEdge_23167053594653
// MI455X (gfx1250) — compile-verified
//
#include <hip/hip_runtime.h>
#include <math.h>

#define N_NODES 50000
#define N_EDGES 600000
#define D 128
#define ND (N_NODES * D)          // 6,400,000 floats per [N,D] buffer
#define MTILES (N_NODES / 16)     // 3125 exact
#define LDS_STRIDE 132            // padded row stride (dwords) -> conflict-free b64 reads

typedef __attribute__((ext_vector_type(2))) float v2f;
typedef __attribute__((ext_vector_type(8))) float v8f;
typedef __attribute__((ext_vector_type(4))) int   v4i;

#define AS1 __attribute__((address_space(1)))
#define AS3 __attribute__((address_space(3)))

#if __has_builtin(__builtin_amdgcn_global_load_async_to_lds_b128) && \
    __has_builtin(__builtin_amdgcn_s_wait_asynccnt)
#define USE_ASYNC_LDS 1
#else
#define USE_ASYNC_LDS 0
#endif

// ---------------------------------------------------------------------------
// float atomic max via int/uint monotone-order trick (init must be -inf)
// ---------------------------------------------------------------------------
__device__ __forceinline__ void atomic_max_f32(float* addr, float val) {
    if (val >= 0.0f) {
        atomicMax((int*)addr, __float_as_int(val));
    } else {
        atomicMin((unsigned int*)addr, __float_as_uint(val));
    }
}

// ---------------------------------------------------------------------------
// ws init: mx = -inf, denom = 0, num = 0
// ---------------------------------------------------------------------------
__global__ void init_ws_kernel(float* __restrict__ mx,
                               float* __restrict__ dn,
                               float* __restrict__ nm) {
    int i = blockIdx.x * blockDim.x + threadIdx.x;
    if (i < ND) {
        mx[i] = -INFINITY;
        dn[i] = 0.0f;
        nm[i] = 0.0f;
    }
}

// ---------------------------------------------------------------------------
// Pass 1: m = node[src] + edge ; atomic max into mx[dst]
// 4 channels per thread (float4 vector loads), 32 threads per edge.
// node_feats (25.6MB) is L2-resident so the gather is cheap.
// ---------------------------------------------------------------------------
__global__ void edge_max_kernel(const float* __restrict__ node,
                                const float* __restrict__ edge,
                                const long long* __restrict__ src,
                                const long long* __restrict__ dst,
                                float* __restrict__ mx) {
    int tid = blockIdx.x * blockDim.x + threadIdx.x;  // < E * 32
    int e = tid >> 5;
    int g = tid & 31;                                 // channel group (4 ch)
    long long s = src[e];
    long long d = dst[e];
    const float4* nf = (const float4*)(node + s * (long long)D);
    const float4* ef = (const float4*)(edge + (long long)e * D);
    float4 a = nf[g];
    float4 b = ef[g];
    float* mp = mx + d * (long long)D + g * 4;
    atomic_max_f32(mp + 0, a.x + b.x);
    atomic_max_f32(mp + 1, a.y + b.y);
    atomic_max_f32(mp + 2, a.z + b.z);
    atomic_max_f32(mp + 3, a.w + b.w);
}

// ---------------------------------------------------------------------------
// Pass 2 (fused): ex = exp(m - mx[dst]); denom += ex; num += m*ex
// (agg = (sum m*ex) / denom with denom constant per (node,channel),
//  so the weighted-sum numerator shares this edge pass)
// ---------------------------------------------------------------------------
__global__ void edge_sum_kernel(const float* __restrict__ node,
                                const float* __restrict__ edge,
                                const long long* __restrict__ src,
                                const long long* __restrict__ dst,
                                const float* __restrict__ mx,
                                float* __restrict__ dn,
                                float* __restrict__ nm) {
    int tid = blockIdx.x * blockDim.x + threadIdx.x;  // < E * 32
    int e = tid >> 5;
    int g = tid & 31;
    long long s = src[e];
    long long d = dst[e];
    const float4* nf = (const float4*)(node + s * (long long)D);
    const float4* ef = (const float4*)(edge + (long long)e * D);
    const float4* mp = (const float4*)(mx + d * (long long)D);
    float4 a = nf[g];
    float4 b = ef[g];
    float4 mxv = mp[g];
    float m0 = a.x + b.x, m1 = a.y + b.y, m2 = a.z + b.z, m3 = a.w + b.w;
    float e0 = __expf(m0 - mxv.x);
    float e1 = __expf(m1 - mxv.y);
    float e2 = __expf(m2 - mxv.z);
    float e3 = __expf(m3 - mxv.w);
    long long base = d * (long long)D + g * 4;
    atomicAdd(dn + base + 0, e0);
    atomicAdd(dn + base + 1, e1);
    atomicAdd(dn + base + 2, e2);
    atomicAdd(dn + base + 3, e3);
    atomicAdd(nm + base + 0, m0 * e0);
    atomicAdd(nm + base + 1, m1 * e1);
    atomicAdd(nm + base + 2, m2 * e2);
    atomicAdd(nm + base + 3, m3 * e3);
}

// ---------------------------------------------------------------------------
// agg = num / denom (0 for isolated nodes), in place into num
// ---------------------------------------------------------------------------
__global__ void agg_kernel(float* __restrict__ nm, const float* __restrict__ dn) {
    int i = blockIdx.x * blockDim.x + threadIdx.x;
    if (i < ND) {
        float d = dn[i];
        nm[i] = (d > 0.0f) ? (nm[i] / d) : 0.0f;
    }
}

// ---------------------------------------------------------------------------
// h = relu(agg @ W^T + b); out = h*scale + node_feats
// - W staged into LDS once per block (async global->LDS when available)
// - A row-fragment preloaded into 64 VGPRs/lane, so the k-loop is
//   ds_load_b64 + v_wmma only
// - one wave per 16-row M-tile; each wave computes all 8 N-tiles
// ---------------------------------------------------------------------------
__global__ void gemm_epilogue_kernel(const float* __restrict__ agg,
                                     const float* __restrict__ W,
                                     const float* __restrict__ bias,
                                     const float* __restrict__ scale,
                                     const float* __restrict__ node,
                                     float* __restrict__ out) {
    __shared__ float lds_w[D * LDS_STRIDE];   // ~66 KB, padded rows

    int tid = threadIdx.x;

    // ---- stage W[128][128] -> lds_w[row*132 + col], 16B chunks ----
#if USE_ASYNC_LDS
#pragma unroll
    for (int i = 0; i < 16; ++i) {
        int c = tid + i * 256;                 // 4096 chunks total
        int row = c >> 5;
        int col = (c & 31) * 4;
        const float* gp = W + row * D + col;
        float* lp = &lds_w[row * LDS_STRIDE + col];
        __builtin_amdgcn_global_load_async_to_lds_b128(
            (AS1 v4i*)gp, (AS3 v4i*)lp, 0, 0);
    }
    __builtin_amdgcn_s_wait_asynccnt(0);
    __syncthreads();
#else
#pragma unroll
    for (int i = 0; i < 16; ++i) {
        int c = tid + i * 256;
        int row = c >> 5;
        int col = (c & 31) * 4;
        *(float4*)&lds_w[row * LDS_STRIDE + col] = *(const float4*)&W[row * D + col];
    }
    __syncthreads();
#endif

    int wave = tid >> 5;
    int lane = tid & 31;
    int mtile = blockIdx.x * (blockDim.x >> 5) + wave;
    if (mtile >= MTILES) return;               // wave-uniform: EXEC stays all-1s

    int half = lane >> 4;                      // 0: lanes 0-15, 1: lanes 16-31
    int l16 = lane & 15;

    // ---- preload A fragments for every k-step (ISA A 16x4 layout) ----
    const float* arow = agg + (long long)(mtile * 16 + l16) * D + half * 2;
    v2f ap[32];
#pragma unroll
    for (int kk = 0; kk < 32; ++kk) {
        ap[kk].x = arow[kk * 4];
        ap[kk].y = arow[kk * 4 + 1];
    }

    // ---- accumulators preloaded with bias (depends on N only) ----
    v8f acc[8];
#pragma unroll
    for (int t = 0; t < 8; ++t) {
        float bv = bias[t * 16 + l16];
        acc[t] = (v8f){bv, bv, bv, bv, bv, bv, bv, bv};
    }

    // ---- k loop: 8 LDS b64 loads, then 8 WMMAs per step ----
    for (int kk = 0; kk < 32; ++kk) {
        int kb = kk * 4 + half * 2;
        v2f bf[8];
#pragma unroll
        for (int t = 0; t < 8; ++t) {
            const float* wrow = &lds_w[(t * 16 + l16) * LDS_STRIDE + kb];
            bf[t].x = wrow[0];                 // B[k][n] = W[n][k]
            bf[t].y = wrow[1];
        }
#pragma unroll
        for (int t = 0; t < 8; ++t) {
            acc[t] = __builtin_amdgcn_wmma_f32_16x16x4_f32(
                false, ap[kk], false, bf[t], (short)0, acc[t], false, false);
        }
    }

    // ---- epilogue: relu, LayerScale, residual ----
#pragma unroll
    for (int t = 0; t < 8; ++t) {
        int n = t * 16 + l16;
        float sc = scale[n];
#pragma unroll
        for (int r = 0; r < 8; ++r) {
            int m = mtile * 16 + r + 8 * half;
            float h = acc[t][r];
            h = (h > 0.0f) ? h : 0.0f;
            long long idx = (long long)m * D + n;
            out[idx] = h * sc + node[idx];
        }
    }
}

// ---------------------------------------------------------------------------
extern "C" void kernel_launch(void* const* d_in, const int* in_sizes, int n_in,
                              void* d_out, int out_size, void* d_ws, size_t ws_size,
                              hipStream_t stream) {
    const float*     node  = (const float*)d_in[0];
    const float*     edge  = (const float*)d_in[1];
    const float*     mlp_w = (const float*)d_in[2];
    const float*     mlp_b = (const float*)d_in[3];
    const float*     scale = (const float*)d_in[4];
    const long long* src   = (const long long*)d_in[5];
    const long long* dst   = (const long long*)d_in[6];
    float* out = (float*)d_out;

    float* ws    = (float*)d_ws;
    float* mx    = ws;            // [N,D]
    float* denom = ws + ND;       // [N,D]
    float* num   = ws + 2 * ND;   // [N,D] -> becomes agg in place

    const int threads = 256;

    // init ws (mx=-inf, denom=num=0)
    init_ws_kernel<<<ND / threads, threads, 0, stream>>>(mx, denom, num);

    // pass 1: segment max
    int edge_blocks = (N_EDGES * 32) / threads;   // 75000
    edge_max_kernel<<<edge_blocks, threads, 0, stream>>>(node, edge, src, dst, mx);

    // pass 2: fused exp-sum (denominator) + weighted-numerator
    edge_sum_kernel<<<edge_blocks, threads, 0, stream>>>(node, edge, src, dst,
                                                         mx, denom, num);

    // agg = num / denom
    agg_kernel<<<ND / threads, threads, 0, stream>>>(num, denom);

    // WMMA GEMM + relu + LayerScale + residual
    int waves_per_block = threads / 32;                        // 8
    int gemm_blocks = (MTILES + waves_per_block - 1) / waves_per_block;  // 391
    gemm_epilogue_kernel<<<gemm_blocks, threads, 0, stream>>>(num, mlp_w, mlp_b,
                                                              scale, node, out);
}